// BLOBLoss_44057774522576
// MI455X (gfx1250) — compile-verified
//
#include <hip/hip_runtime.h>
#include <math.h>

typedef __attribute__((ext_vector_type(2))) float v2f;
typedef __attribute__((ext_vector_type(8))) float v8f;

#define RR 4000
#define CC 20
#define HH 1024
#define WW 1024
#define HS 128
#define WS 128

// workspace layout (float offsets)
#define D_OFF      0
#define D_FLOATS   (CC * HH * WW)            // 20,971,520 floats (80 MB)
#define SC_OFF     (D_OFF + D_FLOATS)        // scores R*C
#define SC_FLOATS  (RR * CC)
#define RM_OFF     (SC_OFF + SC_FLOATS)      // sampled raw M, C*128*128
#define RM_FLOATS  (CC * HS * WS)
#define PM_OFF     (RM_OFF + RM_FLOATS)      // per-strip min/max partials C*64*2
#define PM_FLOATS  (CC * 64 * 2)
#define MM_OFF     (PM_OFF + PM_FLOATS)      // Mmin[C], Mmax[C]
#define MM_FLOATS  (2 * CC)
#define CS_OFF     (MM_OFF + MM_FLOATS)      // per-class loss sums
#define CS_FLOATS  (CC)

// ---------------- scores: avg over K=3 of refine[:, :, 1:], threshold, label mask
__global__ void scores_k(const float* __restrict__ refine,
                         const int* __restrict__ labels,
                         float* __restrict__ ws) {
  int i = blockIdx.x * blockDim.x + threadIdx.x;
  if (i >= RR * CC) return;
  int r = i / CC, c = i % CC;
  const int S = CC + 1;
  float a = refine[0 * RR * S + r * S + c + 1]
          + refine[1 * RR * S + r * S + c + 1]
          + refine[2 * RR * S + r * S + c + 1];
  a *= (1.0f / 3.0f);
  if (a < 0.3f) a = 0.0f;
  if (labels[c] != 1) a = 0.0f;
  ws[SC_OFF + i] = a;
}

// ---------------- corner scatter into per-class delta planes D (1024x1024)
__global__ void scatter_k(const float* __restrict__ rois, float* __restrict__ ws) {
  int i = blockIdx.x * blockDim.x + threadIdx.x;
  if (i >= RR * CC) return;
  int r = i / CC, c = i % CC;
  float s = ws[SC_OFF + i];
  if (s == 0.0f) return;
  int x1 = (int)rois[r * 5 + 1];
  int y1 = (int)rois[r * 5 + 2];
  int x2 = (int)rois[r * 5 + 3];
  int y2 = (int)rois[r * 5 + 4];
  float* Dp = ws + D_OFF + (size_t)c * HH * WW;
  // corners at index >= 1024 only influence discarded rows/cols of M
  if ((unsigned)y1 < HH) {
    if ((unsigned)x1 < WW) atomicAdd(Dp + (size_t)y1 * WW + x1,  s);
    if ((unsigned)x2 < WW) atomicAdd(Dp + (size_t)y1 * WW + x2, -s);
  }
  if ((unsigned)y2 < HH) {
    if ((unsigned)x1 < WW) atomicAdd(Dp + (size_t)y2 * WW + x1, -s);
    if ((unsigned)x2 < WW) atomicAdd(Dp + (size_t)y2 * WW + x2,  s);
  }
}

// ---------------- stage 1: inclusive prefix along x via WMMA  (E = rowscan(D))
// One wave per (class, 16-row strip). P = T(16x16) * U(upper-tri ones) + carry.
__global__ void __launch_bounds__(32) xscan_k(float* __restrict__ ws) {
  int wid = blockIdx.x;
  int c  = wid >> 6;
  int yt = wid & 63;
  int lane = threadIdx.x;
  int half = lane >> 4;       // 0: lanes 0-15, 1: lanes 16-31
  int l16  = lane & 15;
  float* Dp = ws + D_OFF + (size_t)c * HH * WW + (size_t)(yt * 16) * WW;

  // B = U chunks: U[k][n] = (k <= n).  B layout: VGPR0 <-> K{k0, k0+2}, VGPR1 <-> K{k0+1, k0+3}
  int kb = 2 * half;
  v2f b0, b1, b2, b3;
  b0.x = (kb + 0  <= l16) ? 1.f : 0.f;  b0.y = (kb + 1  <= l16) ? 1.f : 0.f;
  b1.x = (kb + 4  <= l16) ? 1.f : 0.f;  b1.y = (kb + 5  <= l16) ? 1.f : 0.f;
  b2.x = (kb + 8  <= l16) ? 1.f : 0.f;  b2.y = (kb + 9  <= l16) ? 1.f : 0.f;
  b3.x = (kb + 12 <= l16) ? 1.f : 0.f;  b3.y = (kb + 13 <= l16) ? 1.f : 0.f;

  v8f carry = {0.f, 0.f, 0.f, 0.f, 0.f, 0.f, 0.f, 0.f};
  int src = half ? 31 : 15;   // lane holding column N=15 for this half's rows

  for (int xt = 0; xt < 64; ++xt) {
    int x0 = xt * 16;
    // A = data tile rows: lane holds T[l16][k0 + 2*half + {0,1}] (aligned float2)
    const float* rp = Dp + (size_t)l16 * WW + x0 + kb;
    v2f a0 = *(const v2f*)(rp + 0);
    v2f a1 = *(const v2f*)(rp + 4);
    v2f a2 = *(const v2f*)(rp + 8);
    v2f a3 = *(const v2f*)(rp + 12);
    v8f d = carry;
    d = __builtin_amdgcn_wmma_f32_16x16x4_f32(false, a0, false, b0, (short)0, d, false, false);
    d = __builtin_amdgcn_wmma_f32_16x16x4_f32(false, a1, false, b1, (short)0, d, false, false);
    d = __builtin_amdgcn_wmma_f32_16x16x4_f32(false, a2, false, b2, (short)0, d, false, false);
    d = __builtin_amdgcn_wmma_f32_16x16x4_f32(false, a3, false, b3, (short)0, d, false, false);
#pragma unroll
    for (int v = 0; v < 8; ++v) {
      Dp[(size_t)(v + half * 8) * WW + x0 + l16] = d[v];   // write E in place
      carry[v] = __shfl(d[v], src, 32);                    // row totals -> next C
    }
  }
}

// ---------------- stage 2: inclusive prefix along y via WMMA (M = colscan(E))
// One wave per (class, 16-col strip). M = L(lower-tri ones) * E + carry_col.
// Harvest min/max and sampled values on the fly; never store full M.
__global__ void __launch_bounds__(32) yscan_k(float* __restrict__ ws) {
  int wid = blockIdx.x;
  int c  = wid >> 6;
  int xs = wid & 63;
  int lane = threadIdx.x;
  int half = lane >> 4;
  int l16  = lane & 15;
  const float* Dp = ws + D_OFF + (size_t)c * HH * WW;
  float* Rm = ws + RM_OFF + (size_t)c * HS * WS;
  int x0 = xs * 16;
  int kb = 2 * half;

  // A = Ltri chunks: Ltri[m][k] = (k <= m); lane holds A[l16][k0 + 2*half + {0,1}]
  v2f a0, a1, a2, a3;
  a0.x = (kb + 0  <= l16) ? 1.f : 0.f;  a0.y = (kb + 1  <= l16) ? 1.f : 0.f;
  a1.x = (kb + 4  <= l16) ? 1.f : 0.f;  a1.y = (kb + 5  <= l16) ? 1.f : 0.f;
  a2.x = (kb + 8  <= l16) ? 1.f : 0.f;  a2.y = (kb + 9  <= l16) ? 1.f : 0.f;
  a3.x = (kb + 12 <= l16) ? 1.f : 0.f;  a3.y = (kb + 13 <= l16) ? 1.f : 0.f;

  float carry = 0.0f;           // per-column carry (column = l16)
  float mn = 3.0e38f, mx = -3.0e38f;

  for (int yt = 0; yt < 64; ++yt) {
    const float* tp = Dp + (size_t)(yt * 16) * WW + x0;
    // B = E rows chunk: b_q.{x,y} = E[k0 + 2*half + {0,1}][x0 + l16]
    v2f b0, b1, b2, b3;
    b0.x = tp[(size_t)(kb + 0)  * WW + l16];  b0.y = tp[(size_t)(kb + 1)  * WW + l16];
    b1.x = tp[(size_t)(kb + 4)  * WW + l16];  b1.y = tp[(size_t)(kb + 5)  * WW + l16];
    b2.x = tp[(size_t)(kb + 8)  * WW + l16];  b2.y = tp[(size_t)(kb + 9)  * WW + l16];
    b3.x = tp[(size_t)(kb + 12) * WW + l16];  b3.y = tp[(size_t)(kb + 13) * WW + l16];
    v8f d = {carry, carry, carry, carry, carry, carry, carry, carry};
    d = __builtin_amdgcn_wmma_f32_16x16x4_f32(false, a0, false, b0, (short)0, d, false, false);
    d = __builtin_amdgcn_wmma_f32_16x16x4_f32(false, a1, false, b1, (short)0, d, false, false);
    d = __builtin_amdgcn_wmma_f32_16x16x4_f32(false, a2, false, b2, (short)0, d, false, false);
    d = __builtin_amdgcn_wmma_f32_16x16x4_f32(false, a3, false, b3, (short)0, d, false, false);
#pragma unroll
    for (int v = 0; v < 8; ++v) { mn = fminf(mn, d[v]); mx = fmaxf(mx, d[v]); }
    // sampled rows of this tile are M=0 (half 0) and M=8 (half 1) -> both in d[0];
    // sampled cols are l16 in {0,8} -> lanes 0,8,16,24
    if ((lane & 7) == 0) {
      Rm[(size_t)(yt * 2 + half) * WS + ((x0 + l16) >> 3)] = d[0];
    }
    carry = __shfl(d[7], 16 + l16, 32);   // last row (M=15) lives in d[7] of lanes 16..31
  }
  for (int off = 16; off; off >>= 1) {
    mn = fminf(mn, __shfl_xor(mn, off, 32));
    mx = fmaxf(mx, __shfl_xor(mx, off, 32));
  }
  if (lane == 0) {
    ws[PM_OFF + (c * 64 + xs) * 2 + 0] = mn;
    ws[PM_OFF + (c * 64 + xs) * 2 + 1] = mx;
  }
}

// ---------------- per-class global min/max
__global__ void minmax_k(float* __restrict__ ws) {
  int c = blockIdx.x;
  int t = threadIdx.x;   // 64 threads
  __shared__ float smn[64], smx[64];
  smn[t] = ws[PM_OFF + (c * 64 + t) * 2 + 0];
  smx[t] = ws[PM_OFF + (c * 64 + t) * 2 + 1];
  __syncthreads();
  for (int s = 32; s; s >>= 1) {
    if (t < s) { smn[t] = fminf(smn[t], smn[t + s]); smx[t] = fmaxf(smx[t], smx[t + s]); }
    __syncthreads();
  }
  if (t == 0) { ws[MM_OFF + c] = smn[0]; ws[MM_OFF + CC + c] = smx[0]; }
}

// ---------------- per-class loss terms (deterministic tree reduction)
__global__ void terms_k(const float* __restrict__ blob,
                        const int* __restrict__ labels,
                        float* __restrict__ ws) {
  int c = blockIdx.x;
  int t = threadIdx.x;   // 128 threads
  const float* B  = blob + (size_t)c * HS * WS;
  const float* Rm = ws + RM_OFF + (size_t)c * HS * WS;
  float mxb = -3.0e38f, myb = -3.0e38f, cM = -3.0e38f, rM = -3.0e38f;
  for (int i = 0; i < HS; ++i) {
    float bc = B[(size_t)i * WS + t];     // column t
    float br = B[(size_t)t * WS + i];     // row t
    bc = fminf(fmaxf(bc, 1e-6f), 1.0f - 1e-6f);
    br = fminf(fmaxf(br, 1e-6f), 1.0f - 1e-6f);
    mxb = fmaxf(mxb, bc);
    myb = fmaxf(myb, br);
    cM  = fmaxf(cM, Rm[(size_t)i * WS + t]);
    rM  = fmaxf(rM, Rm[(size_t)t * WS + i]);
  }
  float vc = 0.f;
  for (int k = 0; k < CC; ++k) vc += (labels[k] == 1) ? 1.f : 0.f;
  float nvc = (float)CC - vc;
  float mmin = ws[MM_OFF + c], mmax = ws[MM_OFF + CC + c];
  float inv = 1.0f / (mmax - mmin + 1e-6f);
  float mxl = (((cM - mmin) * inv) >= 0.5f) ? 1.f : 0.f;
  float myl = (((rM - mmin) * inv) >= 0.5f) ? 1.f : 0.f;
  float vm = (labels[c] == 1) ? 1.f : 0.f;
  float nv = 1.f - vm;
  float term = (-logf(mxb)) * mxl * vm / (vc * (float)WS)
             + (-logf(1.f - mxb)) * nv / (nvc * (float)WS)
             + (-logf(myb)) * myl * vm / (vc * (float)HS)
             + (-logf(1.f - myb)) * nv / (nvc * (float)HS);
  __shared__ float s[128];
  s[t] = term;
  __syncthreads();
  for (int st = 64; st; st >>= 1) {
    if (t < st) s[t] += s[t + st];
    __syncthreads();
  }
  if (t == 0) ws[CS_OFF + c] = s[0];
}

// ---------------- final scalar
__global__ void final_k(const float* __restrict__ ws, float* __restrict__ out) {
  if (blockIdx.x == 0 && threadIdx.x == 0) {
    float s = 0.f;
    for (int c = 0; c < CC; ++c) s += ws[CS_OFF + c];
    out[0] = s;
  }
}

extern "C" void kernel_launch(void* const* d_in, const int* in_sizes, int n_in,
                              void* d_out, int out_size, void* d_ws, size_t ws_size,
                              hipStream_t stream) {
  (void)in_sizes; (void)n_in; (void)out_size; (void)ws_size;
  // d_in order: mil_result(unused), refine_result, blob_conv, rois, labels, H, W
  const float* refine = (const float*)d_in[1];
  const float* blob   = (const float*)d_in[2];
  const float* rois   = (const float*)d_in[3];
  const int*   labels = (const int*)d_in[4];
  float* ws = (float*)d_ws;

  hipMemsetAsync(ws + D_OFF, 0, (size_t)D_FLOATS * sizeof(float), stream);

  int n = RR * CC;
  scores_k <<<(n + 255) / 256, 256, 0, stream>>>(refine, labels, ws);
  scatter_k<<<(n + 255) / 256, 256, 0, stream>>>(rois, ws);
  xscan_k  <<<CC * 64, 32, 0, stream>>>(ws);
  yscan_k  <<<CC * 64, 32, 0, stream>>>(ws);
  minmax_k <<<CC, 64, 0, stream>>>(ws);
  terms_k  <<<CC, 128, 0, stream>>>(blob, labels, ws);
  final_k  <<<1, 32, 0, stream>>>(ws, (float*)d_out);
}